// GRCN_25142738550915
// MI455X (gfx1250) — compile-verified
//
#include <hip/hip_runtime.h>
#include <hip/hip_bf16.h>

typedef __attribute__((ext_vector_type(2))) float v2f;
typedef __attribute__((ext_vector_type(8))) float v8f;

constexpr int N_ = 8192;
constexpr int F_ = 256;
constexpr int H_ = 128;
constexpr int C_ = 16;
constexpr int K_ = 50;

// ---------------- GEMM tile configuration ----------------
#define BM 128
#define BN 64
#define BK 16
#define LDA_S (BM + 4)
#define LDB_S (BN + 4)

// C[M,Nd] = A[M,Kd] @ B   (TB==false: B is Kd x Nd row-major;
//                          TB==true : B is Nd x Kd row-major, used transposed)
// epilogue: v = acc * rowscale[row] (opt) + bias[col] (opt); relu (opt)
template <bool TB>
__global__ void __launch_bounds__(256)
wmma_gemm(const float* __restrict__ A, const float* __restrict__ B,
          float* __restrict__ C, int M, int Nd, int Kd,
          int lda, int ldb, int ldc,
          const float* __restrict__ rowscale,
          const float* __restrict__ bias, int do_relu) {
  __shared__ float As[BK][LDA_S];
  __shared__ float Bs[BK][LDB_S];

  const int tid  = threadIdx.x;
  const int lane = tid & 31;
  const int wave = tid >> 5;   // 0..7
  const int wm   = wave >> 1;  // 0..3 : wave row
  const int wn   = wave & 1;   // 0..1 : wave col
  const int half = lane >> 4;  // 0/1  : lane half
  const int lr   = lane & 15;

  const int row0 = blockIdx.y * BM;
  const int col0 = blockIdx.x * BN;

  v8f acc[2][2] = {};

  for (int k0 = 0; k0 < Kd; k0 += BK) {
    // --- stage A tile (BM x BK) transposed into As[k][m] ---
#pragma unroll
    for (int it = 0; it < (BM * BK) / 256; ++it) {
      int idx = tid + it * 256;
      int m = idx >> 4;
      int k = idx & 15;
      int gr = row0 + m, gk = k0 + k;
      float v = 0.0f;
      if (gr < M && gk < Kd) v = A[(size_t)gr * lda + gk];
      As[k][m] = v;
    }
    // --- stage B tile (BK x BN) into Bs[k][n] ---
#pragma unroll
    for (int it = 0; it < (BK * BN) / 256; ++it) {
      int idx = tid + it * 256;
      float v = 0.0f;
      if (!TB) {
        int k = idx >> 6;
        int n = idx & 63;
        int gk = k0 + k, gn = col0 + n;
        if (gk < Kd && gn < Nd) v = B[(size_t)gk * ldb + gn];
        Bs[k][n] = v;
      } else {
        int n = idx >> 4;
        int k = idx & 15;
        int gk = k0 + k, gn = col0 + n;
        if (gk < Kd && gn < Nd) v = B[(size_t)gn * ldb + gk];
        Bs[k][n] = v;
      }
    }
    __syncthreads();

#pragma unroll
    for (int ks = 0; ks < BK; ks += 4) {
      const int klo = ks + half * 2;  // lanes 0-15 -> K={ks,ks+1}; 16-31 -> K={ks+2,ks+3}
      v2f a0, a1, b0, b1;
      a0.x = As[klo][wm * 32 + lr];       a0.y = As[klo + 1][wm * 32 + lr];
      a1.x = As[klo][wm * 32 + 16 + lr];  a1.y = As[klo + 1][wm * 32 + 16 + lr];
      b0.x = Bs[klo][wn * 32 + lr];       b0.y = Bs[klo + 1][wn * 32 + lr];
      b1.x = Bs[klo][wn * 32 + 16 + lr];  b1.y = Bs[klo + 1][wn * 32 + 16 + lr];
      acc[0][0] = __builtin_amdgcn_wmma_f32_16x16x4_f32(false, a0, false, b0, (short)0, acc[0][0], false, false);
      acc[0][1] = __builtin_amdgcn_wmma_f32_16x16x4_f32(false, a0, false, b1, (short)0, acc[0][1], false, false);
      acc[1][0] = __builtin_amdgcn_wmma_f32_16x16x4_f32(false, a1, false, b0, (short)0, acc[1][0], false, false);
      acc[1][1] = __builtin_amdgcn_wmma_f32_16x16x4_f32(false, a1, false, b1, (short)0, acc[1][1], false, false);
    }
    __syncthreads();
  }

  // --- epilogue: C/D layout — lane<16: rows r, cols lr; lane>=16: rows 8+r ---
#pragma unroll
  for (int mi = 0; mi < 2; ++mi)
#pragma unroll
    for (int ni = 0; ni < 2; ++ni)
#pragma unroll
      for (int r = 0; r < 8; ++r) {
        int gr = row0 + wm * 32 + mi * 16 + half * 8 + r;
        int gc = col0 + wn * 32 + ni * 16 + lr;
        if (gr < M && gc < Nd) {
          float v = acc[mi][ni][r];
          if (rowscale) v *= rowscale[gr];
          if (bias) v += bias[gc];
          if (do_relu) v = fmaxf(v, 0.0f);
          C[(size_t)gr * ldc + gc] = v;
        }
      }
}

// ---------------- row sum -> D^{-1/2} ----------------
__global__ void __launch_bounds__(256)
rowsum_rsqrt_kernel(const float* __restrict__ Mtx, float* __restrict__ dinv, int n) {
  __shared__ float red[256];
  const int row = blockIdx.x;
  const float4* p4 = (const float4*)(Mtx + (size_t)row * n);
  float s = 0.f;
  for (int j = threadIdx.x; j < n / 4; j += 256) {
    float4 v = p4[j];
    s += v.x + v.y + v.z + v.w;
  }
  red[threadIdx.x] = s;
  __syncthreads();
  for (int off = 128; off > 0; off >>= 1) {
    if (threadIdx.x < off) red[threadIdx.x] += red[threadIdx.x + off];
    __syncthreads();
  }
  if (threadIdx.x == 0) {
    float d = red[0];
    dinv[row] = d > 0.f ? rsqrtf(d) : 0.f;
  }
}

// out[i,f] = in[i,f] * rs[i] * (wd ? wd[f] : 1)   (safe in-place)
__global__ void __launch_bounds__(256)
rowcolscale_kernel(const float* __restrict__ in, const float* __restrict__ rs,
                   const float* __restrict__ wd, float* __restrict__ out,
                   int n, int f) {
  int t = blockIdx.x * blockDim.x + threadIdx.x;
  if (t < n * f) {
    int i = t / f;
    int j = t - i * f;
    float v = in[t] * rs[i];
    if (wd) v *= wd[j];
    out[t] = v;
  }
}

// e[i,:] /= max(||e[i,:]||_2, 1e-12), one block per row
__global__ void __launch_bounds__(256)
rownorm_kernel(float* __restrict__ e, int f) {
  __shared__ float red[256];
  const int row = blockIdx.x;
  float* p = e + (size_t)row * f;
  float s = 0.f;
  for (int j = threadIdx.x; j < f; j += 256) {
    float v = p[j];
    s += v * v;
  }
  red[threadIdx.x] = s;
  __syncthreads();
  for (int off = 128; off > 0; off >>= 1) {
    if (threadIdx.x < off) red[threadIdx.x] += red[threadIdx.x + off];
    __syncthreads();
  }
  float inv = 1.0f / fmaxf(sqrtf(red[0]), 1e-12f);
  for (int j = threadIdx.x; j < f; j += 256) p[j] *= inv;
}

// iterative top-K per row; row cached in LDS (8192 f32 = 32 KB of the 320 KB LDS)
__global__ void __launch_bounds__(256)
topk_kernel(const float* __restrict__ S, float* __restrict__ vals,
            int* __restrict__ idx, int n, int kk) {
  __shared__ float sdata[N_];
  __shared__ float rv[256];
  __shared__ int ri[256];
  const int row = blockIdx.x;
  const float* p = S + (size_t)row * n;
  for (int j = threadIdx.x; j < n; j += 256) sdata[j] = p[j];
  __syncthreads();
  for (int t = 0; t < kk; ++t) {
    float best = -3.4e38f;
    int bi = 0;
    for (int j = threadIdx.x; j < n; j += 256) {
      float v = sdata[j];
      if (v > best) { best = v; bi = j; }
    }
    rv[threadIdx.x] = best;
    ri[threadIdx.x] = bi;
    __syncthreads();
    for (int off = 128; off > 0; off >>= 1) {
      if (threadIdx.x < off) {
        float vo = rv[threadIdx.x + off];
        int io = ri[threadIdx.x + off];
        if (vo > rv[threadIdx.x] || (vo == rv[threadIdx.x] && io < ri[threadIdx.x])) {
          rv[threadIdx.x] = vo;
          ri[threadIdx.x] = io;
        }
      }
      __syncthreads();
    }
    if (threadIdx.x == 0) {
      vals[(size_t)row * kk + t] = rv[0];
      idx[(size_t)row * kk + t] = ri[0];
      sdata[ri[0]] = -3.4e38f;
    }
    __syncthreads();
  }
}

// Afinal += Aknn + Aknn^T  (Afinal pre-initialized to Adj)
__global__ void __launch_bounds__(256)
scatter_kernel(const float* __restrict__ vals, const int* __restrict__ idx,
               float* __restrict__ Af, int n, int kk) {
  int t = blockIdx.x * blockDim.x + threadIdx.x;
  if (t < n * kk) {
    int i = t / kk;
    int j = idx[t];
    float v = vals[t];
    atomicAdd(&Af[(size_t)i * n + j], v);
    atomicAdd(&Af[(size_t)j * n + i], v);
  }
}

extern "C" void kernel_launch(void* const* d_in, const int* in_sizes, int n_in,
                              void* d_out, int out_size, void* d_ws, size_t ws_size,
                              hipStream_t stream) {
  const float* x   = (const float*)d_in[0];  // [N,F]
  const float* Adj = (const float*)d_in[1];  // [N,N]
  const float* wd1 = (const float*)d_in[2];  // [F]
  const float* wd2 = (const float*)d_in[3];  // [F]
  const float* W1  = (const float*)d_in[4];  // [F,H]
  const float* b1  = (const float*)d_in[5];  // [H]
  const float* W2  = (const float*)d_in[6];  // [H,C]
  const float* b2  = (const float*)d_in[7];  // [C]
  float* out = (float*)d_out;                // [N,C]

  // workspace layout (floats)
  float* Sbuf  = (float*)d_ws;                    // N*N  (S, then Afinal)
  float* buf1  = Sbuf + (size_t)N_ * N_;          // N*F  (y1, y2)
  float* buf2  = buf1 + (size_t)N_ * F_;          // N*F  (h, e)
  float* buf3  = buf2 + (size_t)N_ * F_;          // N*H  (Z1 / Zs1)
  float* buf4  = buf3 + (size_t)N_ * H_;          // N*H  (t)
  float* buf5  = buf4 + (size_t)N_ * H_;          // N*C  (Z2 / Zs2)
  float* dinv  = buf5 + (size_t)N_ * C_;          // N
  float* dinvF = dinv + N_;                       // N
  float* valsb = dinvF + N_;                      // N*K
  int*   idxb  = (int*)(valsb + (size_t)N_ * K_); // N*K

  const dim3 blk(256);

  // 1) dinv = D^{-1/2} of Adj
  rowsum_rsqrt_kernel<<<N_, blk, 0, stream>>>(Adj, dinv, N_);

  // 2) y1 = dinv[i] * x * wd1   ; h = relu(dinv[i] * (Adj @ y1))
  rowcolscale_kernel<<<(N_ * F_ + 255) / 256, blk, 0, stream>>>(x, dinv, wd1, buf1, N_, F_);
  wmma_gemm<false><<<dim3(F_ / BN, N_ / BM), blk, 0, stream>>>(
      Adj, buf1, buf2, N_, F_, N_, N_, F_, F_, dinv, nullptr, 1);

  // 3) y2 = dinv[i] * h * wd2   ; e = dinv[i] * (Adj @ y2); row-L2-normalize
  rowcolscale_kernel<<<(N_ * F_ + 255) / 256, blk, 0, stream>>>(buf2, dinv, wd2, buf1, N_, F_);
  wmma_gemm<false><<<dim3(F_ / BN, N_ / BM), blk, 0, stream>>>(
      Adj, buf1, buf2, N_, F_, N_, N_, F_, F_, dinv, nullptr, 0);
  rownorm_kernel<<<N_, blk, 0, stream>>>(buf2, F_);

  // 4) S = e @ e^T
  wmma_gemm<true><<<dim3(N_ / BN, N_ / BM), blk, 0, stream>>>(
      buf2, buf2, Sbuf, N_, N_, F_, F_, F_, N_, nullptr, nullptr, 0);

  // 5) top-K per row
  topk_kernel<<<N_, blk, 0, stream>>>(Sbuf, valsb, idxb, N_, K_);

  // 6) Afinal = Adj + Aknn + Aknn^T  (reuse S buffer)
  hipMemcpyAsync(Sbuf, Adj, (size_t)N_ * N_ * sizeof(float),
                 hipMemcpyDeviceToDevice, stream);
  scatter_kernel<<<(N_ * K_ + 255) / 256, blk, 0, stream>>>(valsb, idxb, Sbuf, N_, K_);

  // 7) dinvF = D^{-1/2} of Afinal
  rowsum_rsqrt_kernel<<<N_, blk, 0, stream>>>(Sbuf, dinvF, N_);

  // 8) Z1 = x @ W1 ; Zs1 = dinvF[j]*Z1 ; t = relu(dinvF[i]*(Afinal@Zs1) + b1)
  wmma_gemm<false><<<dim3(H_ / BN, N_ / BM), blk, 0, stream>>>(
      x, W1, buf3, N_, H_, F_, F_, H_, H_, nullptr, nullptr, 0);
  rowcolscale_kernel<<<(N_ * H_ + 255) / 256, blk, 0, stream>>>(buf3, dinvF, nullptr, buf3, N_, H_);
  wmma_gemm<false><<<dim3(H_ / BN, N_ / BM), blk, 0, stream>>>(
      Sbuf, buf3, buf4, N_, H_, N_, N_, H_, H_, dinvF, b1, 1);

  // 9) Z2 = t @ W2 ; Zs2 = dinvF[j]*Z2 ; out = dinvF[i]*(Afinal@Zs2) + b2
  wmma_gemm<false><<<dim3((C_ + BN - 1) / BN, N_ / BM), blk, 0, stream>>>(
      buf4, W2, buf5, N_, C_, H_, H_, C_, C_, nullptr, nullptr, 0);
  rowcolscale_kernel<<<(N_ * C_ + 255) / 256, blk, 0, stream>>>(buf5, dinvF, nullptr, buf5, N_, C_);
  wmma_gemm<false><<<dim3((C_ + BN - 1) / BN, N_ / BM), blk, 0, stream>>>(
      Sbuf, buf5, out, N_, C_, N_, N_, C_, C_, dinvF, b2, 0);

  (void)in_sizes; (void)n_in; (void)out_size; (void)ws_size;
}